// GATLayer_53326313947259
// MI455X (gfx1250) — compile-verified
//
#include <hip/hip_runtime.h>
#include <hip/hip_bf16.h>

typedef float v2f __attribute__((ext_vector_type(2)));
typedef float v8f __attribute__((ext_vector_type(8)));

#define IN_DIM   128
#define HEADS    8
#define OUT_DIM  32
#define HC       256   // HEADS*OUT_DIM
#define NEG_SLOPE 0.2f

// ---------- helpers ----------
__device__ __forceinline__ unsigned encf(float f) {
    unsigned b = __float_as_uint(f);
    return ((int)b >= 0) ? (b | 0x80000000u) : ~b;
}
__device__ __forceinline__ float decf(unsigned u) {
    return (u & 0x80000000u) ? __uint_as_float(u ^ 0x80000000u)
                             : __uint_as_float(~u);
}
__device__ __forceinline__ void atomic_add_f32(float* p, float v) {
    __hip_atomic_fetch_add(p, v, __ATOMIC_RELAXED, __HIP_MEMORY_SCOPE_AGENT);
}
__device__ __forceinline__ void atomic_max_u32(unsigned* p, unsigned v) {
    __hip_atomic_fetch_max(p, v, __ATOMIC_RELAXED, __HIP_MEMORY_SCOPE_AGENT);
}

// ---------- 0: init m(enc)= -inf floor, denom=0, out=bias ----------
__global__ void init_buffers(unsigned* __restrict__ mEnc, float* __restrict__ denom,
                             float* __restrict__ out, const float* __restrict__ bias,
                             int nh, int nout) {
    int t = blockIdx.x * blockDim.x + threadIdx.x;
    if (t < nh) { mEnc[t] = 0u; denom[t] = 0.0f; }
    if (t < nout) out[t] = bias[t & (OUT_DIM - 1)];
}

// ---------- 1: h = x @ W via V_WMMA_F32_16X16X4_F32 ----------
// grid = N/16 blocks, 256 threads (8 waves). Wave w computes rows
// [blk*16, blk*16+16) x cols [w*32, w*32+32): two 16x16 tiles sharing A.
__global__ void __launch_bounds__(256) gemm_h_wmma(const float* __restrict__ x,
                                                   const float* __restrict__ W,
                                                   float* __restrict__ h) {
    const int wave = threadIdx.x >> 5;
    const int lane = threadIdx.x & 31;
    const int l16  = lane & 15;
    const int half = lane >> 4;            // 0: K pair {0,1}, 1: K pair {2,3}
    const int rowTile = blockIdx.x;

    const int row  = rowTile * 16 + l16;   // A-matrix row for this lane
    const int col0 = wave * 32 + l16;      // B/D column, tile 0
    const int col1 = col0 + 16;            // tile 1

    const float* xRow = x + (size_t)row * IN_DIM + half * 2;
    const float* Wofs = W + (size_t)(half * 2) * HC;

    v8f acc0 = {};
    v8f acc1 = {};
#pragma unroll 4
    for (int kb = 0; kb < IN_DIM; kb += 4) {
        v2f a = *(const v2f*)(xRow + kb);                 // A[m][kb+2h .. +1]
        const float* wk = Wofs + (size_t)kb * HC;
        v2f b0, b1;
        b0.x = wk[col0];       b0.y = wk[HC + col0];      // B[kb+2h][n], B[kb+2h+1][n]
        b1.x = wk[col1];       b1.y = wk[HC + col1];
        acc0 = __builtin_amdgcn_wmma_f32_16x16x4_f32(false, a, false, b0,
                                                     (short)0, acc0, false, false);
        acc1 = __builtin_amdgcn_wmma_f32_16x16x4_f32(false, a, false, b1,
                                                     (short)0, acc1, false, false);
    }
    // D layout: VGPR r -> M = r + half*8, N = l16
#pragma unroll
    for (int r = 0; r < 8; ++r) {
        int m = rowTile * 16 + half * 8 + r;
        h[(size_t)m * HC + col0] = acc0[r];
        h[(size_t)m * HC + col1] = acc1[r];
    }
}

// ---------- 2: a_src/a_dst = <h[n,h,:], att_{src,dst}[h,:]> ----------
__global__ void att_dots(const float* __restrict__ h,
                         const float* __restrict__ att_src,
                         const float* __restrict__ att_dst,
                         float* __restrict__ a_src, float* __restrict__ a_dst,
                         int total /* N*HEADS */) {
    int t = blockIdx.x * blockDim.x + threadIdx.x;
    if (t >= total) return;
    int hh = t & (HEADS - 1);
    const float4* hp = (const float4*)(h + (size_t)t * OUT_DIM); // n*256 + hh*32 == t*32
    const float4* as = (const float4*)(att_src + hh * OUT_DIM);
    const float4* ad = (const float4*)(att_dst + hh * OUT_DIM);
    float s = 0.f, d = 0.f;
#pragma unroll
    for (int i = 0; i < OUT_DIM / 4; ++i) {
        float4 hv = hp[i], a = as[i], b = ad[i];
        s += hv.x * a.x + hv.y * a.y + hv.z * a.z + hv.w * a.w;
        d += hv.x * b.x + hv.y * b.y + hv.z * b.z + hv.w * b.w;
    }
    a_src[t] = s;
    a_dst[t] = d;
}

// ---------- 3: segment-max of leaky_relu(a_src[s]+a_dst[d]) over dst ----------
__global__ void edge_max(const int* __restrict__ ei, int E_, int N_,
                         const float* __restrict__ a_src, const float* __restrict__ a_dst,
                         unsigned* __restrict__ mEnc) {
    int t = blockIdx.x * blockDim.x + threadIdx.x;
    int total = (E_ + N_) * HEADS;
    if (t >= total) return;
    int e = t >> 3, hh = t & (HEADS - 1);
    int s, d;
    if (e < E_) { s = ei[e]; d = ei[E_ + e]; } else { s = d = e - E_; }
    float v = a_src[s * HEADS + hh] + a_dst[d * HEADS + hh];
    v = v > 0.f ? v : NEG_SLOPE * v;
    atomic_max_u32(&mEnc[d * HEADS + hh], encf(v));
}

// ---------- 4: denom[d,h] += exp(e - m[d,h]) ----------
__global__ void edge_expsum(const int* __restrict__ ei, int E_, int N_,
                            const float* __restrict__ a_src, const float* __restrict__ a_dst,
                            const unsigned* __restrict__ mEnc, float* __restrict__ denom) {
    int t = blockIdx.x * blockDim.x + threadIdx.x;
    int total = (E_ + N_) * HEADS;
    if (t >= total) return;
    int e = t >> 3, hh = t & (HEADS - 1);
    int s, d;
    if (e < E_) { s = ei[e]; d = ei[E_ + e]; } else { s = d = e - E_; }
    float v = a_src[s * HEADS + hh] + a_dst[d * HEADS + hh];
    v = v > 0.f ? v : NEG_SLOPE * v;
    float ex = __expf(v - decf(mEnc[d * HEADS + hh]));
    atomic_add_f32(&denom[d * HEADS + hh], ex);
}

// ---------- 5: one wave per edge; lane = output channel (OUT_DIM==32==wave32) ----
__global__ void __launch_bounds__(256) edge_aggregate(const int* __restrict__ ei, int E_, int N_,
                                                      const float* __restrict__ a_src,
                                                      const float* __restrict__ a_dst,
                                                      const unsigned* __restrict__ mEnc,
                                                      const float* __restrict__ denom,
                                                      const float* __restrict__ h,
                                                      float* __restrict__ out) {
    int waveId = (int)((blockIdx.x * blockDim.x + threadIdx.x) >> 5);
    int lane = threadIdx.x & 31;
    if (waveId >= E_ + N_) return;          // uniform across the wave
    int e = waveId;
    int s, d;
    if (e < E_) { s = ei[e]; d = ei[E_ + e]; } else { s = d = e - E_; }
    // lanes 0..7 (replicated x4) each own one head's alpha
    int hh = lane & (HEADS - 1);
    float v = a_src[s * HEADS + hh] + a_dst[d * HEADS + hh];
    v = v > 0.f ? v : NEG_SLOPE * v;
    float alpha = __expf(v - decf(mEnc[d * HEADS + hh])) / denom[d * HEADS + hh];

    const float* hrow = h + (size_t)s * HC;
    float acc = 0.f;
#pragma unroll
    for (int k = 0; k < HEADS; ++k) {
        float ak = __shfl(alpha, k, 32);                 // broadcast head k's alpha
        acc += ak * hrow[k * OUT_DIM + lane];            // coalesced 128B per head
    }
    atomic_add_f32(&out[(size_t)d * OUT_DIM + lane], acc * (1.0f / HEADS));
}

// ---------- launcher ----------
extern "C" void kernel_launch(void* const* d_in, const int* in_sizes, int n_in,
                              void* d_out, int out_size, void* d_ws, size_t ws_size,
                              hipStream_t stream) {
    const float* x       = (const float*)d_in[0];
    const int*   ei      = (const int*)d_in[1];   // edge_index flattened [2*E]
    /* d_in[2] = batch (unused) */
    const float* W       = (const float*)d_in[3];
    const float* att_src = (const float*)d_in[4];
    const float* att_dst = (const float*)d_in[5];
    const float* bias    = (const float*)d_in[6];
    float*       out     = (float*)d_out;

    const int N_ = in_sizes[0] / IN_DIM;          // 50000
    const int E_ = in_sizes[1] / 2;               // 800000
    const int NH = N_ * HEADS;

    // workspace layout (floats)
    char* ws = (char*)d_ws;
    float*    h     = (float*)ws;                         ws += (size_t)N_ * HC * sizeof(float);
    float*    aS    = (float*)ws;                         ws += (size_t)NH * sizeof(float);
    float*    aD    = (float*)ws;                         ws += (size_t)NH * sizeof(float);
    unsigned* mEnc  = (unsigned*)ws;                      ws += (size_t)NH * sizeof(unsigned);
    float*    denom = (float*)ws;

    const int BS = 256;

    // 0) init accumulators (out = bias, m = floor, denom = 0)
    {
        int nout = N_ * OUT_DIM;
        int mx = nout > NH ? nout : NH;
        init_buffers<<<(mx + BS - 1) / BS, BS, 0, stream>>>(mEnc, denom, out, bias, NH, nout);
    }
    // 1) h = x @ W  (fp32 WMMA)
    gemm_h_wmma<<<N_ / 16, BS, 0, stream>>>(x, W, h);
    // 2) attention dot products
    att_dots<<<(NH + BS - 1) / BS, BS, 0, stream>>>(h, att_src, att_dst, aS, aD, NH);
    // 3) scatter max
    {
        int total = (E_ + N_) * HEADS;
        edge_max<<<(total + BS - 1) / BS, BS, 0, stream>>>(ei, E_, N_, aS, aD, mEnc);
    }
    // 4) exp + scatter sum
    {
        int total = (E_ + N_) * HEADS;
        edge_expsum<<<(total + BS - 1) / BS, BS, 0, stream>>>(ei, E_, N_, aS, aD, mEnc, denom);
    }
    // 5) weighted aggregation, head-mean folded in (out pre-seeded with bias)
    {
        long long threads = (long long)(E_ + N_) * 32;
        int blocks = (int)((threads + BS - 1) / BS);
        edge_aggregate<<<blocks, BS, 0, stream>>>(ei, E_, N_, aS, aD, mEnc, denom, h, out);
    }
}